// InbucketPoolingLayer_12627203851166
// MI455X (gfx1250) — compile-verified
//
#include <hip/hip_runtime.h>
#include <hip/hip_bf16.h>

// InbucketPoolingLayer for MI455X (gfx1250, wave32).
// Pipeline (all deterministic, all on `stream`):
//   1) init + per-instance bbox via order-preserving-uint atomicMin/Max
//   2) 20-bit voxel key per point
//   3) stable 3-pass LSD radix sort (8-bit digits), ballot-based wave32 ranks
//   4) unpool_ind + coord pair-mean
//   5) feature pair-mean: async global->LDS gather + chained V_WMMA_F32_16X16X4_F32
//   6) seps // 2
//
// Workspace use (u32 units): 2*B*3 + 4*N + 256*ceil(N/4096) + 2048
// (~16.3 MB for N = 1,048,576).

typedef float v2f __attribute__((ext_vector_type(2)));
typedef float v8f __attribute__((ext_vector_type(8)));
typedef int v4i_ __attribute__((vector_size(16)));
typedef __attribute__((address_space(1))) v4i_* as1_v4i_ptr;
typedef __attribute__((address_space(3))) v4i_* as3_v4i_ptr;

#define RADIX_EPB 4096   // elements per block in radix kernels
#define SCAN_EPB  2048   // elements per block in scan kernels

// ---------- helpers ----------
__device__ __forceinline__ unsigned encF(float f) {
  unsigned u = __float_as_uint(f);
  return (u & 0x80000000u) ? ~u : (u | 0x80000000u);   // order-preserving
}
__device__ __forceinline__ float decF(unsigned e) {
  unsigned u = (e & 0x80000000u) ? (e ^ 0x80000000u) : ~e;
  return __uint_as_float(u);
}
__device__ __forceinline__ int inst_of(int i, const int* __restrict__ seps, int Bn) {
  // searchsorted(seps, i, side='right') == #{k : seps[k] <= i}
  int inst = 0;
  for (int k = 0; k < Bn; ++k) inst += (i >= seps[k]) ? 1 : 0;
  return (inst < Bn) ? inst : (Bn - 1);
}

// 16B async copy global -> LDS (ASYNCcnt-tracked; bypasses VGPRs).
// Builtin signature (from hipcc diagnostic): param0 = v4i addrspace(1)*, param1 = LDS side.
__device__ __forceinline__ void async_copy16(const void* g, void* l) {
#if __has_builtin(__builtin_amdgcn_global_load_async_to_lds_b128)
  __builtin_amdgcn_global_load_async_to_lds_b128(
      (as1_v4i_ptr)(uintptr_t)g, (as3_v4i_ptr)(unsigned)(uintptr_t)l, 0, 0);
#else
  unsigned loff = (unsigned)(uintptr_t)l;   // generic LDS addr low 32 bits == LDS offset
  asm volatile("global_load_async_to_lds_b128 %0, %1, off" ::"v"(loff), "v"(g) : "memory");
#endif
}
__device__ __forceinline__ void async_wait0() {
#if __has_builtin(__builtin_amdgcn_s_wait_asynccnt)
  __builtin_amdgcn_s_wait_asynccnt(0);
#else
  asm volatile("s_wait_asynccnt 0x0" ::: "memory");
#endif
}

// ---------- 1) bbox ----------
__global__ void init_bbox(unsigned* bmin, unsigned* bmax, int n) {
  for (int i = threadIdx.x; i < n; i += blockDim.x) { bmin[i] = 0xFFFFFFFFu; bmax[i] = 0u; }
}

__global__ __launch_bounds__(256) void bbox_kernel(const float* __restrict__ coords,
                                                   const int* __restrict__ seps, int Bn, int N,
                                                   unsigned* __restrict__ bmin,
                                                   unsigned* __restrict__ bmax) {
  int i = blockIdx.x * blockDim.x + threadIdx.x;
  unsigned act = __builtin_amdgcn_ballot_w32(i < N);
  int inst = (i < N) ? inst_of(i, seps, Bn) : 0;
  unsigned enc[3];
  for (int k = 0; k < 3; ++k) enc[k] = (i < N) ? encF(coords[(size_t)i * 3 + k]) : 0u;

  bool fast = false;
  if (act == 0xFFFFFFFFu) {
    int i0 = __builtin_amdgcn_readfirstlane(inst);
    if (__builtin_amdgcn_ballot_w32(inst == i0) == 0xFFFFFFFFu) {
      fast = true;                               // wave-uniform instance: reduce in-wave first
      for (int k = 0; k < 3; ++k) {
        unsigned mn = enc[k], mx = enc[k];
        for (int off = 16; off; off >>= 1) {
          unsigned a = (unsigned)__shfl_xor((int)mn, off, 32);
          unsigned b = (unsigned)__shfl_xor((int)mx, off, 32);
          mn = (a < mn) ? a : mn;
          mx = (b > mx) ? b : mx;
        }
        if ((threadIdx.x & 31) == 0) {
          atomicMin(&bmin[i0 * 3 + k], mn);
          atomicMax(&bmax[i0 * 3 + k], mx);
        }
      }
    }
  }
  if (!fast && i < N) {
    for (int k = 0; k < 3; ++k) {
      atomicMin(&bmin[inst * 3 + k], enc[k]);
      atomicMax(&bmax[inst * 3 + k], enc[k]);
    }
  }
}

// ---------- 2) keys ----------
__global__ __launch_bounds__(256) void key_kernel(const float* __restrict__ coords,
                                                  const int* __restrict__ seps, int Bn, int N,
                                                  const unsigned* __restrict__ bmin,
                                                  const unsigned* __restrict__ bmax,
                                                  unsigned* __restrict__ keys,
                                                  unsigned* __restrict__ idx) {
  int i = blockIdx.x * blockDim.x + threadIdx.x;
  if (i >= N) return;
  int inst = inst_of(i, seps, Bn);
  int q[3];
  for (int k = 0; k < 3; ++k) {
    float lo = decF(bmin[inst * 3 + k]);
    float hi = decF(bmax[inst * 3 + k]);
    float u = (coords[(size_t)i * 3 + k] - lo) / fmaxf(hi - lo, 1e-12f);
    int t = (int)(u * 40.0f);                      // truncation, matches astype(int32) for u>=0
    q[k] = t < 0 ? 0 : (t > 39 ? 39 : t);
  }
  keys[i] = (unsigned)(inst * 64000 + (q[0] * 40 + q[1]) * 40 + q[2]);
  idx[i] = (unsigned)i;
}

// ---------- 3) radix sort ----------
__global__ __launch_bounds__(256) void radix_hist(const unsigned* __restrict__ keys, int N,
                                                  int shift, unsigned* __restrict__ counts,
                                                  int NB) {
  __shared__ unsigned h[256];
  int t = threadIdx.x;
  h[t] = 0;
  __syncthreads();
  int base = blockIdx.x * RADIX_EPB;
  for (int c = 0; c < RADIX_EPB / 256; ++c) {
    int i = base + c * 256 + t;
    if (i < N) atomicAdd(&h[(keys[i] >> shift) & 0xFFu], 1u);
  }
  __syncthreads();
  counts[(size_t)t * NB + blockIdx.x] = h[t];      // digit-major layout for the global scan
}

// generic exclusive scan, level 1 (also serves as level 2 with partials==nullptr)
__global__ __launch_bounds__(256) void scan_l1(unsigned* __restrict__ data, int M,
                                               unsigned* __restrict__ partials) {
  __shared__ unsigned s[256];
  int t = threadIdx.x;
  int base = blockIdx.x * SCAN_EPB + t * 8;
  unsigned v[8], tsum = 0;
  for (int k = 0; k < 8; ++k) {
    unsigned x = (base + k < M) ? data[base + k] : 0u;
    v[k] = tsum;                                   // exclusive within thread
    tsum += x;
  }
  s[t] = tsum;
  __syncthreads();
  for (int off = 1; off < 256; off <<= 1) {        // Hillis-Steele inclusive
    unsigned x = (t >= off) ? s[t - off] : 0u;
    __syncthreads();
    s[t] += x;
    __syncthreads();
  }
  unsigned excl = s[t] - tsum;
  for (int k = 0; k < 8; ++k)
    if (base + k < M) data[base + k] = excl + v[k];
  if (t == 255 && partials) partials[blockIdx.x] = s[255];
}

__global__ __launch_bounds__(256) void scan_l3(unsigned* __restrict__ data, int M,
                                               const unsigned* __restrict__ partials) {
  int base = blockIdx.x * SCAN_EPB + threadIdx.x * 8;
  unsigned p = partials[blockIdx.x];
  for (int k = 0; k < 8; ++k)
    if (base + k < M) data[base + k] += p;
}

// stable scatter: ballot-based wave ranks + cross-wave prefix + per-block running totals
__global__ __launch_bounds__(256) void radix_scatter(const unsigned* __restrict__ keysIn,
                                                     const unsigned* __restrict__ idxIn, int N,
                                                     int shift,
                                                     const unsigned* __restrict__ scanned, int NB,
                                                     unsigned* __restrict__ keysOut,
                                                     unsigned* __restrict__ idxOut) {
  __shared__ unsigned waveHist[8][256];
  __shared__ unsigned running[256];
  const int t = threadIdx.x, lane = t & 31, wv = t >> 5;
  running[t] = 0;
  const int base = blockIdx.x * RADIX_EPB;
  for (int c = 0; c < RADIX_EPB / 256; ++c) {
    for (int k = t; k < 8 * 256; k += 256) ((unsigned*)waveHist)[k] = 0;
    __syncthreads();

    int i = base + c * 256 + t;
    bool valid = (i < N);
    unsigned key = valid ? keysIn[i] : 0u;
    unsigned id = valid ? idxIn[i] : 0u;
    unsigned d = (key >> shift) & 0xFFu;

    unsigned peers = __builtin_amdgcn_ballot_w32(valid);
#pragma unroll
    for (int b = 0; b < 8; ++b) {
      unsigned bal = __builtin_amdgcn_ballot_w32(((d >> b) & 1u) != 0u);
      peers &= ((d >> b) & 1u) ? bal : ~bal;
    }
    unsigned waveRank = __popc(peers & ((1u << lane) - 1u));
    if (valid && waveRank == 0) waveHist[wv][d] = __popc(peers);
    __syncthreads();

    if (valid) {
      unsigned prefix = 0;
      for (int w = 0; w < wv; ++w) prefix += waveHist[w][d];
      unsigned gbase = scanned[(size_t)d * NB + blockIdx.x];
      unsigned pos = gbase + running[d] + prefix + waveRank;
      keysOut[pos] = key;
      idxOut[pos] = id;
    }
    __syncthreads();
    unsigned tot = 0;
#pragma unroll
    for (int w = 0; w < 8; ++w) tot += waveHist[w][t];
    running[t] += tot;
    __syncthreads();
  }
}

// ---------- 4) unpool indices + coord pair-means ----------
__global__ __launch_bounds__(256) void unpool_coord_kernel(const float* __restrict__ coords,
                                                           const unsigned* __restrict__ sortedIdx,
                                                           int N, float* __restrict__ outCoord,
                                                           int* __restrict__ outUnpool) {
  int r = blockIdx.x * blockDim.x + threadIdx.x;
  if (r >= N) return;
  unsigned si = sortedIdx[r];
  outUnpool[si] = r >> 1;
  if (!(r & 1) && (r + 1) < N) {
    int s = r >> 1;
    unsigned sj = sortedIdx[r + 1];
    for (int k = 0; k < 3; ++k)
      outCoord[(size_t)s * 3 + k] =
          0.5f * (coords[(size_t)si * 3 + k] + coords[(size_t)sj * 3 + k]);
  }
}

// ---------- 5) feature pair-means via async->LDS gather + V_WMMA_F32_16X16X4_F32 ----------
// Tile: 16 subbuckets x 128 features per wave (one wave per block).
// 32 gathered rows async-copied into LDS (stride 136 floats -> conflict-free B reads),
// 16B per lane per row, ASYNCcnt-tracked, no VGPR staging.
// D(16x16) = sum_{j=0..7} A_j(16x4) x B_j(4x16); A_j[m,k]=0.5 iff (4j+k)>>1==m.
// Exact fp32: 0.5*x is exact; single final rounding == (x+y)*0.5.
__global__ __launch_bounds__(32) void reduce_feat_wmma(const float* __restrict__ feat,
                                                       const unsigned* __restrict__ sortedIdx,
                                                       float* __restrict__ outFeat, int N,
                                                       int RN) {
  __shared__ float tile[32][136];
  const int lane = threadIdx.x;
  const int sbBase = blockIdx.x * 16;
  const int rowBase = sbBase * 2;

  if (sbBase + 16 <= RN) {                         // full tile: WMMA path (wave-uniform branch)
    int mySrc = (int)sortedIdx[rowBase + lane];
#pragma unroll 4
    for (int r = 0; r < 32; ++r) {
      unsigned s = (unsigned)__shfl(mySrc, r, 32);
      async_copy16(feat + (size_t)s * 128 + lane * 4, &tile[r][lane * 4]);
    }
    async_wait0();
    __syncthreads();

    const int h = lane >> 4, n = lane & 15;
    for (int f0 = 0; f0 < 128; f0 += 16) {
      v8f acc = {};
#pragma unroll
      for (int j = 0; j < 8; ++j) {
        // A layout (16x4 f32): lane -> M=n, K in {2h, 2h+1}; both slots share the condition
        float aval = (n == (2 * j + h)) ? 0.5f : 0.0f;
        v2f A = {aval, aval};
        // B layout (4x16 f32): lane -> N=n, K in {2h, 2h+1} -> sorted rows 4j+2h, 4j+2h+1
        int r0 = 4 * j + 2 * h;
        v2f Bv = {tile[r0][f0 + n], tile[r0 + 1][f0 + n]};
        acc = __builtin_amdgcn_wmma_f32_16x16x4_f32(false, A, false, Bv, (short)0, acc, false,
                                                    false);
      }
#pragma unroll
      for (int v = 0; v < 8; ++v)                  // D: VGPR v, half h -> subbucket v + 8h
        outFeat[(size_t)(sbBase + v + 8 * h) * 128 + f0 + n] = acc[v];
    }
  } else {                                         // ragged tail: scalar path
    for (int s = sbBase; s < RN; ++s) {
      unsigned i0 = sortedIdx[2 * s], i1 = sortedIdx[2 * s + 1];
      for (int c = lane * 4; c < lane * 4 + 4; ++c)
        outFeat[(size_t)s * 128 + c] =
            0.5f * (feat[(size_t)i0 * 128 + c] + feat[(size_t)i1 * 128 + c]);
    }
  }
}

// ---------- 6) seps // 2 ----------
__global__ void sep_kernel(const int* __restrict__ seps, int Bn, int* __restrict__ outSep) {
  int i = blockIdx.x * blockDim.x + threadIdx.x;
  if (i < Bn) outSep[i] = seps[i] / 2;
}

// ---------- host ----------
extern "C" void kernel_launch(void* const* d_in, const int* in_sizes, int n_in, void* d_out,
                              int out_size, void* d_ws, size_t ws_size, hipStream_t stream) {
  const float* coords = (const float*)d_in[0];
  const float* feat = (const float*)d_in[1];
  const int* seps = (const int*)d_in[2];
  const int N = in_sizes[0] / 3;
  const int Bn = in_sizes[2];
  const int RN = N / 2;

  // d_out: reduced_coord (RN*3 f32) | reduced_sep (Bn i32) | reduced_feat (RN*128 f32)
  //        | unpool_ind (N i32), concatenated in return order.
  float* outCoord = (float*)d_out;
  int* outSep = (int*)(outCoord + (size_t)RN * 3);
  float* outFeat = (float*)(outSep + Bn);
  int* outUnpool = (int*)(outFeat + (size_t)RN * 128);

  unsigned* ws = (unsigned*)d_ws;
  unsigned* bmin = ws;          ws += (size_t)Bn * 3;
  unsigned* bmax = ws;          ws += (size_t)Bn * 3;
  unsigned* keysA = ws;         ws += N;
  unsigned* idxA = ws;          ws += N;
  unsigned* keysB = ws;         ws += N;
  unsigned* idxB = ws;          ws += N;
  const int NB = (N + RADIX_EPB - 1) / RADIX_EPB;
  unsigned* counts = ws;        ws += (size_t)256 * NB;
  unsigned* partials = ws;      ws += 2048;

  const int gN = (N + 255) / 256;

  init_bbox<<<1, 256, 0, stream>>>(bmin, bmax, Bn * 3);
  bbox_kernel<<<gN, 256, 0, stream>>>(coords, seps, Bn, N, bmin, bmax);
  key_kernel<<<gN, 256, 0, stream>>>(coords, seps, Bn, N, bmin, bmax, keysA, idxA);

  // 3 stable LSD passes over the 20-bit key: shifts 0, 8, 16. A->B->A->B (result in B).
  unsigned* sk[4] = {keysA, keysB, keysA, keysB};
  unsigned* si[4] = {idxA, idxB, idxA, idxB};
  const int M = 256 * NB;
  const int G1 = (M + SCAN_EPB - 1) / SCAN_EPB;
  for (int p = 0; p < 3; ++p) {
    radix_hist<<<NB, 256, 0, stream>>>(sk[p], N, 8 * p, counts, NB);
    scan_l1<<<G1, 256, 0, stream>>>(counts, M, partials);
    scan_l1<<<1, 256, 0, stream>>>(partials, G1, (unsigned*)nullptr);
    scan_l3<<<G1, 256, 0, stream>>>(counts, M, partials);
    radix_scatter<<<NB, 256, 0, stream>>>(sk[p], si[p], N, 8 * p, counts, NB, sk[p + 1],
                                          si[p + 1]);
  }
  unsigned* sortedIdx = si[3];

  unpool_coord_kernel<<<gN, 256, 0, stream>>>(coords, sortedIdx, N, outCoord, outUnpool);
  reduce_feat_wmma<<<(RN + 15) / 16, 32, 0, stream>>>(feat, sortedIdx, outFeat, N, RN);
  sep_kernel<<<(Bn + 255) / 256, 256, 0, stream>>>(seps, Bn, outSep);
}